// FrameAlignedPointError_52063593562812
// MI455X (gfx1250) — compile-verified
//
#include <hip/hip_runtime.h>

typedef __attribute__((ext_vector_type(2))) float v2f;
typedef __attribute__((ext_vector_type(8))) float v8f;

#define BATCHES 4

// Workspace layout (floats): accP[BATCHES][64] then accF[BATCHES][64].
// acc[r*8 + c] holds D[row r][col c] of the 16x16 WMMA accumulator (rows/cols 0..7 used).

__global__ void fape_zero_acc(float* ws) {
    int i = blockIdx.x * blockDim.x + threadIdx.x;
    if (i < BATCHES * 128) ws[i] = 0.0f;
}

// Point-side statistics: rows of the tall-skinny matrix are [x0 x1 x2 | y0 y1 y2 | 1].
// Gram via V_WMMA_F32_16X16X4_F32 gives Sxx, Sxy, Syy, sx, sy, N in one accumulator.
__global__ void fape_point_stats(const float* __restrict__ x,
                                 const float* __restrict__ y,
                                 float* __restrict__ accP, int N) {
    const int b      = blockIdx.x;
    const int lane   = threadIdx.x & 31;
    // Wave id is wave-uniform: force it into an SGPR so the chunk loop is a
    // uniform scalar loop (EXEC stays all-1s across every WMMA, per ISA req).
    const int wave   = __builtin_amdgcn_readfirstlane((int)(threadIdx.x >> 5));
    const int nwaves = blockDim.x >> 5;
    const int gwave  = blockIdx.y * nwaves + wave;      // global wave id within batch
    const int stride = gridDim.y * nwaves;
    const int effL   = lane & 15;                       // matrix row/col index (M or N)
    const int hi     = lane >> 4;                       // lanes 16-31 supply K=2,3

    // Per-lane source pointer: rows 0-2 -> x coords, 3-5 -> y coords, 6 -> ones row.
    const float* base = (effL < 3) ? (x + (size_t)b * N * 3 + effL)
                      : (effL < 6) ? (y + (size_t)b * N * 3 + (effL - 3))
                                   : (x + (size_t)b * N * 3);   // dummy (scaled by 0)
    const float scale = (effL < 6) ? 1.0f : 0.0f;
    const float bias  = (effL == 6) ? 1.0f : 0.0f;

    v8f c = {0.f, 0.f, 0.f, 0.f, 0.f, 0.f, 0.f, 0.f};
    const int nchunks = N >> 2;                         // K=4 points per WMMA
    for (int ch = gwave; ch < nchunks; ch += stride) {
        const int p0 = ch * 4 + hi * 2;
        v2f ab;
        ab.x = base[(size_t)p0 * 3] * scale + bias;         // K = 0 (or 2)
        ab.y = base[(size_t)(p0 + 1) * 3] * scale + bias;   // K = 1 (or 3)
        c = __builtin_amdgcn_wmma_f32_16x16x4_f32(false, ab, false, ab,
                                                  (short)0, c, false, false);
    }
    // C/D layout: lanes 0-15 hold rows 0..7 (vgpr index = row), col = lane.
    if (lane < 8) {
        float* acc = accP + b * 64;
#pragma unroll
        for (int r = 0; r < 8; ++r)
            atomicAdd(&acc[r * 8 + lane], c[r]);
    }
}

// Frame-side statistics: rows of Q are indexed by t=(l,i); Q[t, j] = R[l, j, i].
// Rows are [Qx0 Qx1 Qx2 | Qy0 Qy1 Qy2]; Gram gives Σ RxRxᵀ, Σ RxRyᵀ, Σ RyRyᵀ.
__global__ void fape_frame_stats(const float* __restrict__ Rx,
                                 const float* __restrict__ Ry,
                                 float* __restrict__ accF, int N) {
    const int b      = blockIdx.x;
    const int lane   = threadIdx.x & 31;
    const int wave   = __builtin_amdgcn_readfirstlane((int)(threadIdx.x >> 5));
    const int nwaves = blockDim.x >> 5;
    const int gwave  = blockIdx.y * nwaves + wave;
    const int stride = gridDim.y * nwaves;
    const int effL   = lane & 15;
    const int hi     = lane >> 4;

    const float* base = (effL < 3) ? (Rx + (size_t)b * N * 9)
                                   : (Ry + (size_t)b * N * 9);
    const int   j     = (effL < 3) ? effL : ((effL < 6) ? (effL - 3) : 0);
    const float scale = (effL < 6) ? 1.0f : 0.0f;

    v8f c = {0.f, 0.f, 0.f, 0.f, 0.f, 0.f, 0.f, 0.f};
    const int nchunks = (N * 3) >> 2;                   // 3N rows of Q, 4 per WMMA
    for (int ch = gwave; ch < nchunks; ch += stride) {
        const int t0 = ch * 4 + hi * 2;
        const int l0 = t0 / 3, i0 = t0 - l0 * 3;
        const int t1 = t0 + 1;
        const int l1 = t1 / 3, i1 = t1 - l1 * 3;
        v2f ab;
        ab.x = base[(size_t)l0 * 9 + j * 3 + i0] * scale;
        ab.y = base[(size_t)l1 * 9 + j * 3 + i1] * scale;
        c = __builtin_amdgcn_wmma_f32_16x16x4_f32(false, ab, false, ab,
                                                  (short)0, c, false, false);
    }
    if (lane < 8) {
        float* acc = accF + b * 64;
#pragma unroll
        for (int r = 0; r < 8; ++r)
            atomicAdd(&acc[r * 8 + lane], c[r]);
    }
}

// Final 3x3 contractions in double precision; writes the scalar mean.
__global__ void fape_finalize(const float* __restrict__ accP,
                              const float* __restrict__ accF,
                              float* __restrict__ out, int N, double inv_count) {
    if (threadIdx.x != 0 || blockIdx.x != 0) return;
    double total = 0.0;
    const double invN = 1.0 / (double)N;
    for (int b = 0; b < BATCHES; ++b) {
        const float* P = accP + b * 64;
        const float* F = accF + b * 64;
        double sx[3], sy[3];
        for (int j = 0; j < 3; ++j) { sx[j] = P[6 * 8 + j]; sy[j] = P[6 * 8 + 3 + j]; }
        for (int j = 0; j < 3; ++j) {
            for (int m = 0; m < 3; ++m) {
                const double Gx  = (double)P[j * 8 + m]           - sx[j] * sx[m] * invN;
                const double Mxy = (double)P[j * 8 + 3 + m]       - sx[j] * sy[m] * invN;
                const double Gy  = (double)P[(j + 3) * 8 + 3 + m] - sy[j] * sy[m] * invN;
                const double SA  = (double)F[j * 8 + m];
                const double SC  = (double)F[j * 8 + 3 + m];
                const double SB  = (double)F[(j + 3) * 8 + 3 + m];
                total += SA * Gx - 2.0 * SC * Mxy + SB * Gy;
            }
        }
    }
    *out = (float)(total * inv_count);
}

extern "C" void kernel_launch(void* const* d_in, const int* in_sizes, int n_in,
                              void* d_out, int out_size, void* d_ws, size_t ws_size,
                              hipStream_t stream) {
    const float* x  = (const float*)d_in[0];
    const float* Rx = (const float*)d_in[1];
    const float* y  = (const float*)d_in[2];
    const float* Ry = (const float*)d_in[3];

    const int BN = in_sizes[0] / 3;      // B*N
    const int N  = BN / BATCHES;         // 2048

    float* ws   = (float*)d_ws;
    float* accP = ws;
    float* accF = ws + BATCHES * 64;

    fape_zero_acc<<<1, 512, 0, stream>>>(ws);

    dim3 grid(BATCHES, 4);
    fape_point_stats<<<grid, 256, 0, stream>>>(x, y, accP, N);
    fape_frame_stats<<<grid, 256, 0, stream>>>(Rx, Ry, accF, N);

    const double inv_count = 1.0 / ((double)BATCHES * (double)N * (double)N * 3.0);
    fape_finalize<<<1, 32, 0, stream>>>(accP, accF, (float*)d_out, N, inv_count);
}